// LinearAccInterpolation_56229711839313
// MI455X (gfx1250) — compile-verified
//
#include <hip/hip_runtime.h>
#include <stdint.h>

// LinearAccInterpolation on MI455X (gfx1250).
// Memory-bound op (~111 MB traffic, ~0.23 GFLOP): optimized via CDNA5
// async global->LDS staging (ASYNCcnt path) for coalesced input reads,
// wave32-friendly 256-thread blocks, one batch per thread.

#define BLK 256
#define MAX_KP 16   // harness uses N_KP = 11

__device__ __forceinline__ void async_ld_b128(uint32_t lds_addr, const void* gaddr) {
  // GLOBAL_LOAD_ASYNC_TO_LDS_B128: VDST = LDS byte address, VADDR = 64-bit global address
  asm volatile("global_load_async_to_lds_b128 %0, %1, off"
               :: "v"(lds_addr), "v"(gaddr) : "memory");
}

__device__ __forceinline__ void async_ld_b64(uint32_t lds_addr, const void* gaddr) {
  asm volatile("global_load_async_to_lds_b64 %0, %1, off"
               :: "v"(lds_addr), "v"(gaddr) : "memory");
}

__device__ __forceinline__ void wait_async0() {
  asm volatile("s_wait_asynccnt 0x0" ::: "memory");
}

__global__ __launch_bounds__(BLK)
void LinearAccInterpolation_kernel(const int* __restrict__ idx,
                                   const float* __restrict__ value,
                                   const float* __restrict__ speed,
                                   const float* __restrict__ acc,
                                   float* __restrict__ out,
                                   int B, int n_kp, int R) {
  __shared__ float s_val[BLK * MAX_KP * 2];  // 32 KB max (22 KB used at n_kp=11)
  __shared__ float s_v0[BLK * 2];            // 2 KB
  __shared__ float s_a0[BLK * 2];            // 2 KB
  __shared__ int   s_idx[MAX_KP];

  const int tid        = threadIdx.x;
  const int blockStart = blockIdx.x * BLK;
  const int nb         = min(BLK, B - blockStart);

  if (tid < n_kp && tid < MAX_KP) s_idx[tid] = idx[tid];

  if (nb > 0) {
    // --- stage value[blockStart : blockStart+nb] into LDS, coalesced b128 ---
    const char*   gval     = (const char*)(value + (size_t)blockStart * n_kp * 2);
    const uint32_t lds_val = (uint32_t)(size_t)(void*)s_val;
    const int totalFloats  = nb * n_kp * 2;    // always even
    const int nf4          = totalFloats >> 2;
    for (int j = tid; j < nf4; j += BLK)
      async_ld_b128(lds_val + (uint32_t)j * 16u, gval + (size_t)j * 16);
    if ((totalFloats & 3) != 0 && tid == 0)    // possible 2-float tail
      async_ld_b64(lds_val + (uint32_t)nf4 * 16u, gval + (size_t)nf4 * 16);

    // --- stage init_speed / init_acc (one float2 per batch) ---
    if (tid < nb) {
      async_ld_b64((uint32_t)(size_t)(void*)s_v0 + (uint32_t)tid * 8u,
                   (const void*)(speed + (size_t)(blockStart + tid) * 2));
      async_ld_b64((uint32_t)(size_t)(void*)s_a0 + (uint32_t)tid * 8u,
                   (const void*)(acc + (size_t)(blockStart + tid) * 2));
    }
  }
  wait_async0();     // this wave's async LDS writes complete
  __syncthreads();   // cross-wave visibility

  if (tid < nb) {
    const int b = blockStart + tid;
    const float* vp = &s_val[tid * n_kp * 2];  // conflict-free: 22-word lane stride
    float vx = s_v0[tid * 2 + 0], vy = s_v0[tid * 2 + 1];
    float ax = s_a0[tid * 2 + 0], ay = s_a0[tid * 2 + 1];
    float y0x = vp[0], y0y = vp[1];
    float2* outp = (float2*)out + (size_t)b * R;
    int row = 0;

    for (int i = 0; i + 1 < n_kp; ++i) {
      const float y1x = vp[(i + 1) * 2 + 0];
      const float y1y = vp[(i + 1) * 2 + 1];
      const float dyx = y1x - y0x;
      const float dyy = y1y - y0y;
      const int n = s_idx[i + 1] - s_idx[i];

      if (n <= 1) {
        outp[row++] = make_float2(y1x, y1y);
        const float nvx = dyx, nvy = dyy;
        ax = nvx - vx;  ay = nvy - vy;
        vx = nvx;       vy = nvy;
      } else {
        const float fn = (float)n;
        const float c1 = fn * (fn + 1.0f) * 0.5f;                  // n(n+1)/2
        const float c2 = fn * (fn + 1.0f) * (fn + 2.0f) / 6.0f;    // n(n+1)(n+2)/6
        const float dax = (dyx - fn * vx - c1 * ax) / c2;
        const float day = (dyy - fn * vy - c1 * ay) / c2;
        for (int k = 1; k <= n; ++k) {
          const float fk = (float)k;
          const float t2 = fk * (fk + 1.0f) * 0.5f;
          const float t3 = fk * (fk + 1.0f) * (fk + 2.0f) / 6.0f;
          const float px = y0x + fk * vx + t2 * ax + t3 * dax;
          const float py = y0y + fk * vy + t2 * ay + t3 * day;
          outp[row++] = make_float2(px, py);
        }
        vx = vx + fn * ax + c1 * dax;
        vy = vy + fn * ay + c1 * day;
        ax = ax + fn * dax;
        ay = ay + fn * day;
      }
      y0x = y1x;  y0y = y1y;
    }
  }
}

extern "C" void kernel_launch(void* const* d_in, const int* in_sizes, int n_in,
                              void* d_out, int out_size, void* d_ws, size_t ws_size,
                              hipStream_t stream) {
  (void)n_in; (void)d_ws; (void)ws_size;
  const int*   idx   = (const int*)d_in[0];     // index: (N_KP,) int32
  const float* value = (const float*)d_in[1];   // value: (B, N_KP, DIM) f32
  const float* speed = (const float*)d_in[2];   // init_speed: (B, 1, DIM) f32
  const float* acc   = (const float*)d_in[3];   // init_acc:   (B, 1, DIM) f32
  float* out = (float*)d_out;

  const int n_kp = in_sizes[0];                 // 11
  const int B    = in_sizes[2] / 2;             // 262144
  const int R    = out_size / (B * 2);          // 40 total output rows per batch

  const int grid = (B + BLK - 1) / BLK;
  hipLaunchKernelGGL(LinearAccInterpolation_kernel, dim3(grid), dim3(BLK), 0, stream,
                     idx, value, speed, acc, out, B, n_kp, R);
}